// MultiplicativeAttention_22574348108155
// MI455X (gfx1250) — compile-verified
//
#include <hip/hip_runtime.h>
#include <hip/hip_bf16.h>

// Problem constants (from reference setup_inputs)
#define BDIM 64
#define TDIM 4096
#define QS   1024
#define KSZ  1024

typedef __attribute__((ext_vector_type(2))) float v2f;
typedef __attribute__((ext_vector_type(4))) float f4;
typedef __attribute__((ext_vector_type(8))) float v8f;

// ---------------------------------------------------------------------------
// Kernel 1: mids[b][k] = sum_q W[k][q] * query[b][q]
// One wave (32 lanes) per 16(k) x 16(b) output tile, K-loop in steps of 4
// using V_WMMA_F32_16X16X4_F32 (f32 A/B/C/D, matches reference precision).
//
// ISA fragment layouts (CDNA5 ISA):
//   A (16x4 f32):  lanes 0-15: M=lane, VGPR0/1 = K0,K1 ; lanes 16-31: K2,K3
//   B (4x16 f32):  lanes 0-15: N=lane, VGPR0/1 = K0,K1 ; lanes 16-31: K2,K3
//   D (16x16 f32): lanes 0-15: N=lane, VGPR r = M=r ; lanes 16-31: M=8+r
// ---------------------------------------------------------------------------
__global__ void mids_wmma_kernel(const float* __restrict__ W,
                                 const float* __restrict__ query,
                                 float* __restrict__ mids /* [BDIM][KSZ] */) {
    const int km   = blockIdx.x;          // 0..63 -> k0 = km*16
    const int bn   = blockIdx.y;          // 0..3  -> b0 = bn*16
    const int lane = threadIdx.x;         // 0..31 (wave32)
    const int half = lane >> 4;           // 0: K={0,1}, 1: K={2,3}
    const int l16  = lane & 15;

    const int k0 = km * 16;
    const int b0 = bn * 16;

    // A fragment source: row (k0+l16) of W, starting at k-offset half*2
    const float* wrow = W + (size_t)(k0 + l16) * QS + half * 2;
    // B fragment source: B[k][n] = query[b0+n][q0+k]
    const float* qrow = query + (size_t)(b0 + l16) * QS + half * 2;

    v8f acc = {};
    for (int q0 = 0; q0 < QS; q0 += 4) {
        v2f a = *(const v2f*)(wrow + q0);   // 8B aligned: q0%4==0, half*2 offset
        v2f b = *(const v2f*)(qrow + q0);
        // 8 args: (neg_a, A, neg_b, B, c_mod, C, reuse_a, reuse_b)
        acc = __builtin_amdgcn_wmma_f32_16x16x4_f32(
            false, a, false, b, (short)0, acc, false, false);
    }

    // D store: lane holds column b = b0+l16; VGPR r -> k = k0 + half*8 + r
    float* dst = mids + (size_t)(b0 + l16) * KSZ + k0 + half * 8;
#pragma unroll
    for (int r = 0; r < 8; ++r) dst[r] = acc[r];
}

// ---------------------------------------------------------------------------
// Kernel 2: scores[b][t] = tanh(key[b][t] . mids[b] + bias)
// Memory-bound streaming of key (1.07 GB, read once -> non-temporal loads).
// Block = 256 threads (8 waves), handles 64 rows of one batch.
// mids[b] (4 KB) staged in LDS, reused by all 64 rows of the block.
// One wave per row: 8 x float4 NT loads per lane, then wave32 xor-reduction.
// ---------------------------------------------------------------------------
__global__ void scores_kernel(const float* __restrict__ key,
                              const float* __restrict__ mids,
                              const float* __restrict__ bias,
                              float* __restrict__ scores /* [BDIM][TDIM] */) {
    __shared__ float smids[KSZ];

    const int b      = blockIdx.x;        // batch
    const int tchunk = blockIdx.y;        // 0..63, 64 rows per block
    const int tid    = threadIdx.x;       // 0..255

    // Cooperative load of mids[b] into LDS: 256 threads x 1 float4 = 1024 f32
    ((f4*)smids)[tid] = ((const f4*)(mids + (size_t)b * KSZ))[tid];
    __syncthreads();

    const int lane = tid & 31;
    const int wave = tid >> 5;            // 0..7
    const float bv = bias[0];

#pragma unroll 1
    for (int r = 0; r < 8; ++r) {
        const int t = tchunk * 64 + wave * 8 + r;
        const f4* krow = (const f4*)(key + ((size_t)b * TDIM + t) * KSZ);
        float acc = 0.f;
#pragma unroll
        for (int i = 0; i < 8; ++i) {
            // 32 lanes x 16B = 512B contiguous per iteration; NT: bypass cache
            f4 kv = __builtin_nontemporal_load(&krow[i * 32 + lane]);
            f4 mv = ((const f4*)smids)[i * 32 + lane];
            acc += kv.x * mv.x + kv.y * mv.y + kv.z * mv.z + kv.w * mv.w;
        }
        // wave32 reduction
#pragma unroll
        for (int off = 16; off > 0; off >>= 1)
            acc += __shfl_xor(acc, off, 32);
        if (lane == 0)
            scores[(size_t)b * TDIM + t] = tanhf(acc + bv);
    }
}

// ---------------------------------------------------------------------------
// Kernel 3: masked softmax over T per batch.
// One block (256 threads) per batch; 16 elements per thread (strided,
// coalesced); LDS tree reductions for max and sum.
// ---------------------------------------------------------------------------
__global__ void softmax_kernel(const float* __restrict__ scores,
                               const float* __restrict__ mask,
                               float* __restrict__ out /* [BDIM][TDIM] */) {
    __shared__ float red[256];
    const int b   = blockIdx.x;
    const int tid = threadIdx.x;

    float vals[16];
    float m = -3.402823466e38f;
#pragma unroll
    for (int j = 0; j < 16; ++j) {
        vals[j] = scores[(size_t)b * TDIM + tid + j * 256];
        m = fmaxf(m, vals[j]);
    }
    red[tid] = m;
    __syncthreads();
    for (int s = 128; s > 0; s >>= 1) {
        if (tid < s) red[tid] = fmaxf(red[tid], red[tid + s]);
        __syncthreads();
    }
    m = red[0];
    __syncthreads();

    float sum = 0.f;
#pragma unroll
    for (int j = 0; j < 16; ++j) {
        float e = expf(vals[j] - m) * mask[(size_t)b * TDIM + tid + j * 256];
        vals[j] = e;
        sum += e;
    }
    red[tid] = sum;
    __syncthreads();
    for (int s = 128; s > 0; s >>= 1) {
        if (tid < s) red[tid] += red[tid + s];
        __syncthreads();
    }
    const float inv = 1.f / red[0];
#pragma unroll
    for (int j = 0; j < 16; ++j)
        out[(size_t)b * TDIM + tid + j * 256] = vals[j] * inv;
}

// ---------------------------------------------------------------------------
// Launcher. Inputs (setup_inputs order): query, key, mask, W, bias. All f32.
// Workspace layout: mids [64*1024] f32, then scores [64*4096] f32 (1.25 MB).
// ---------------------------------------------------------------------------
extern "C" void kernel_launch(void* const* d_in, const int* in_sizes, int n_in,
                              void* d_out, int out_size, void* d_ws, size_t ws_size,
                              hipStream_t stream) {
    const float* query = (const float*)d_in[0];
    const float* key   = (const float*)d_in[1];
    const float* mask  = (const float*)d_in[2];
    const float* W     = (const float*)d_in[3];
    const float* bias  = (const float*)d_in[4];
    float* out = (float*)d_out;

    float* mids   = (float*)d_ws;                  // BDIM*KSZ floats
    float* scores = mids + (size_t)BDIM * KSZ;     // BDIM*TDIM floats

    // K1: 64 (k-tiles) x 4 (b-tiles) waves, one wave per 16x16 tile
    mids_wmma_kernel<<<dim3(KSZ / 16, BDIM / 16), 32, 0, stream>>>(W, query, mids);

    // K2: 64 batches x 64 t-chunks = 4096 blocks of 256 threads
    scores_kernel<<<dim3(BDIM, TDIM / 64), 256, 0, stream>>>(key, mids, bias, scores);

    // K3: one block per batch
    softmax_kernel<<<BDIM, 256, 0, stream>>>(scores, mask, out);
}